// VectorP1FunctionSpace_5342939316636
// MI455X (gfx1250) — compile-verified
//
#include <hip/hip_runtime.h>

typedef float v2f __attribute__((ext_vector_type(2)));
typedef float v8f __attribute__((ext_vector_type(8)));

#define KINC 6  // incident cells per vertex (padded; padded slots have c=1e9)

__device__ __forceinline__ float relu_min6(float a, float b, float c,
                                           float d, float e, float f) {
    // min and relu commute (relu monotone): basis = max(min_k d_k, 0)
    float m = fminf(fminf(fminf(a, b), fminf(c, d)), fminf(e, f));
    return fmaxf(m, 0.0f);
}

__global__ __launch_bounds__(256)
void hat_basis_wmma_kernel(const float* __restrict__ xg,
                           const float* __restrict__ Wg,
                           const float* __restrict__ Cg,
                           const float* __restrict__ wxg,
                           const float* __restrict__ wyg,
                           float* __restrict__ outg,
                           int V, int npoints, int ngroups) {
    extern __shared__ float smem[];
    float* sWC  = smem;             // [0, V*12): W pairs (Wx,Wy); [V*12, V*18): c
    float* sWxy = sWC + V * 18;     // V*2: interleaved (wx[v], wy[v])

    const int tid = threadIdx.x, nthr = blockDim.x;
    // vectorized staging: W as b128, c as b64
    {
        const float4* Wg4 = reinterpret_cast<const float4*>(Wg);   // V*12 % 4 == 0
        float4* sW4 = reinterpret_cast<float4*>(sWC);
        for (int i = tid; i < V * 3; i += nthr) sW4[i] = Wg4[i];
        const float2* Cg2 = reinterpret_cast<const float2*>(Cg);   // V*6 % 2 == 0
        float2* sC2 = reinterpret_cast<float2*>(sWC + V * 12);
        for (int i = tid; i < V * 3; i += nthr) sC2[i] = Cg2[i];
        for (int v = tid; v < V; v += nthr) {
            sWxy[2 * v]     = wxg[v];
            sWxy[2 * v + 1] = wyg[v];
        }
    }
    __syncthreads();

    const int lane  = tid & 31;
    const int wave  = tid >> 5;
    const int group = blockIdx.x * (nthr >> 5) + wave;
    if (group >= ngroups) return;   // wave-uniform exit keeps EXEC full for WMMA

    const int  n  = lane & 15;
    const bool hi = lane >= 16;     // A: supplies K=2,3 (c,junk). D: holds rows 8-15.

    int p  = group * 16 + n;
    int pl = p < npoints ? p : npoints - 1;
    float px = xg[2 * pl], py = xg[2 * pl + 1];
    // B (4x16): K0=x, K1=y (lanes 0-15); K2=1, K3=0 (lanes 16-31).
    // => dot = Wx*x + Wy*y + c*1 + junk*0  (junk finite, so exactly 0 contribution)
    v2f bmat;
    bmat[0] = hi ? 1.0f : px;
    bmat[1] = hi ? 0.0f : py;

    // Row packing: batch of 3 tiles = 48 rows = 8 vertices x 6 cells.
    // Slot s = t*8 + (row&7); vertex-in-batch = (row>=8 ? 4 : 0) + s/6; k = s%6.
    const int row = lane & 15;      // A row this lane feeds (both K-halves same row)
    const int rl  = row & 7;
    const int up4 = (row >> 3) * 4; // rows 8-15 -> vertex set +4 (upper D half)
    // per-lane float-index bases into sWC and constant batch stride
    int af[3];
    const int astride = hi ? (8 * KINC) : (2 * 8 * KINC);   // 48 : 96 floats/batch
#pragma unroll
    for (int t = 0; t < 3; ++t) {
        int s    = t * 8 + rl;
        int jj   = s / 6;
        int kk   = s - jj * 6;
        int idx0 = (up4 + jj) * KINC + kk;      // (v,k) flat index at vbase=0
        af[t]    = hi ? (V * 12 + idx0) : (2 * idx0);
    }
    int wa = hi ? 8 : 0;            // sWxy float index base; +16 per batch

    float accx = 0.0f, accy = 0.0f;
    const int NBfull = V / 8;       // 136 clamp-free batches (covers v in [0, 1088))
    const v8f  zero  = {};

#pragma unroll 2
    for (int vb = 0; vb < NBfull; ++vb) {
        v8f d[3];
#pragma unroll
        for (int t = 0; t < 3; ++t) {
            float a0 = sWC[af[t]];
            float a1 = sWC[af[t] + 1];          // hi lanes: finite junk * 0
            af[t] += astride;
            v2f amat; amat[0] = a0; amat[1] = a1;
            d[t] = __builtin_amdgcn_wmma_f32_16x16x4_f32(
                false, amat, false, bmat, (short)0, zero, false, false);
        }
        // slot s -> (tile s>>3, vgpr s&7); lanes 0-15 give vtx j, lanes 16-31 vtx j+4
        float b0 = relu_min6(d[0][0], d[0][1], d[0][2], d[0][3], d[0][4], d[0][5]);
        float b1 = relu_min6(d[0][6], d[0][7], d[1][0], d[1][1], d[1][2], d[1][3]);
        float b2 = relu_min6(d[1][4], d[1][5], d[1][6], d[1][7], d[2][0], d[2][1]);
        float b3 = relu_min6(d[2][2], d[2][3], d[2][4], d[2][5], d[2][6], d[2][7]);
        accx = fmaf(b0, sWxy[wa + 0], accx);  accy = fmaf(b0, sWxy[wa + 1], accy);
        accx = fmaf(b1, sWxy[wa + 2], accx);  accy = fmaf(b1, sWxy[wa + 3], accy);
        accx = fmaf(b2, sWxy[wa + 4], accx);  accy = fmaf(b2, sWxy[wa + 5], accy);
        accx = fmaf(b3, sWxy[wa + 6], accx);  accy = fmaf(b3, sWxy[wa + 7], accy);
        wa += 16;
    }

    // Fold upper-half vertex partials into lower half: SWAPX16 via ds_swizzle
    float ox = accx + __int_as_float(
        __builtin_amdgcn_ds_swizzle(__float_as_int(accx), 0x401F));
    float oy = accy + __int_as_float(
        __builtin_amdgcn_ds_swizzle(__float_as_int(accy), 0x401F));

    // Scalar epilogue for leftover vertices [NBfull*8, V) -- after the fold so
    // each is counted exactly once. Runs on all lanes; only lanes<16 store.
    for (int v = NBfull * 8; v < V; ++v) {
        float dmin = 3.0e38f;
#pragma unroll
        for (int k = 0; k < KINC; ++k) {
            int idx = v * KINC + k;
            float dk = fmaf(sWC[2 * idx], px,
                       fmaf(sWC[2 * idx + 1], py, sWC[V * 12 + idx]));
            dmin = fminf(dmin, dk);
        }
        float bas = fmaxf(dmin, 0.0f);
        ox = fmaf(bas, sWxy[2 * v],     ox);
        oy = fmaf(bas, sWxy[2 * v + 1], oy);
    }

    if (!hi && p < npoints) {
        float2 o; o.x = ox; o.y = oy;
        *reinterpret_cast<float2*>(outg + 2 * p) = o;
    }
}

extern "C" void kernel_launch(void* const* d_in, const int* in_sizes, int n_in,
                              void* d_out, int out_size, void* d_ws, size_t ws_size,
                              hipStream_t stream) {
    const float* x  = (const float*)d_in[0];
    const float* W  = (const float*)d_in[1];
    const float* C  = (const float*)d_in[2];
    const float* wx = (const float*)d_in[3];
    const float* wy = (const float*)d_in[4];
    float* out = (float*)d_out;

    int V       = in_sizes[3];            // 1089 vertices
    int npoints = in_sizes[0] / 2;        // B*N = 16384
    int ngroups = (npoints + 15) / 16;    // 1024 point-groups (one per wave)

    int threads       = 256;              // 8 wave32 per block
    int wavesPerBlock = threads / 32;
    int blocks        = (ngroups + wavesPerBlock - 1) / wavesPerBlock;
    size_t shmem      = (size_t)(V * 18 + V * 2) * sizeof(float);  // ~87 KB

    hat_basis_wmma_kernel<<<blocks, threads, shmem, stream>>>(
        x, W, C, wx, wy, out, V, npoints, ngroups);
}